// get_model_61993557950588
// MI455X (gfx1250) — compile-verified
//
#include <hip/hip_runtime.h>
#include <math.h>

// ---------------------------------------------------------------------------
// PointNet++ forward for MI455X / gfx1250.
// All MLP layers use V_WMMA_F32_16X16X4_F32, branch-free uniform K loops,
// K padded to multiple of 8, dual accumulator chains, software-pipelined
// loads (next step's fragments fetched before current step's wmmas).
// ---------------------------------------------------------------------------

typedef float v2f __attribute__((ext_vector_type(2)));
typedef float v8f __attribute__((ext_vector_type(8)));

#define B_SZ 16
#define N0   4096

struct LayerP { const float *W, *b, *g, *be; };
struct SAPp   { const float *W0, *b0, *W1, *b1, *W2, *b2; };

// ------------------------- input split / transpose -------------------------
__global__ void split_input(const float* __restrict__ in,
                            float* __restrict__ xyz0, float* __restrict__ pts0) {
  int i = blockIdx.x * blockDim.x + threadIdx.x;      // over B*N0
  if (i >= B_SZ * N0) return;
  int b = i >> 12, n = i & (N0 - 1);
  const float* src = in + (size_t)b * 9 * N0;
  for (int c = 0; c < 3; ++c) xyz0[(size_t)i * 3 + c] = src[c * N0 + n];
  for (int c = 0; c < 6; ++c) pts0[(size_t)i * 6 + c] = src[(3 + c) * N0 + n];
}

// --------------------------- weight/bias packing ---------------------------
// Wp[n*Cinp+k] = (k<Cin) ? W[n*Cin+k]*g[n]*inv : 0      (BN scale folded)
// bp[n]        = b[n]*g[n]*inv + be[n]                  (BN shift folded)
__global__ void pack_weight(const float* __restrict__ W, const float* __restrict__ g,
                            float* __restrict__ Wp, int Cin, int Cinp, int total,
                            float inv) {
  int e = blockIdx.x * blockDim.x + threadIdx.x;
  if (e >= total) return;
  int n = e / Cinp, k = e % Cinp;
  Wp[e] = (k < Cin) ? W[(size_t)n * Cin + k] * (g[n] * inv) : 0.0f;
}

__global__ void pack_bias(const float* __restrict__ b, const float* __restrict__ g,
                          const float* __restrict__ be, float* __restrict__ bp,
                          int Cout, float inv) {
  int n = blockIdx.x * blockDim.x + threadIdx.x;
  if (n >= Cout) return;
  bp[n] = b[n] * (g[n] * inv) + be[n];
}

__global__ void zero_pad_cols(float* __restrict__ X, int ld, int c0, int rows) {
  int w = ld - c0;
  int i = blockIdx.x * blockDim.x + threadIdx.x;
  if (i >= rows * w) return;
  X[(size_t)(i / w) * ld + c0 + (i % w)] = 0.0f;
}

// ------------------------------ FPS (serial) -------------------------------
__global__ __launch_bounds__(256) void fps_kernel(const float* __restrict__ xyz,
                                                  int* __restrict__ fidx,
                                                  int N, int S) {
  __shared__ float dist[4096];
  __shared__ float rv[256];
  __shared__ int   ri[256];
  __shared__ int   cur;
  int b = blockIdx.x, t = threadIdx.x;
  if (t == 0) cur = 0;
  for (int i = t; i < N; i += 256) dist[i] = 1e10f;
  __syncthreads();
  const float* base = xyz + (size_t)b * N * 3;
  for (int s = 0; s < S; ++s) {
    int c = cur;
    if (t == 0) fidx[b * S + s] = c;
    float cx = base[c * 3 + 0], cy = base[c * 3 + 1], cz = base[c * 3 + 2];
    float bv = -1.0f; int bi = 0;
    for (int i = t; i < N; i += 256) {
      float dx = base[i * 3 + 0] - cx;
      float dy = base[i * 3 + 1] - cy;
      float dz = base[i * 3 + 2] - cz;
      float d = dx * dx + dy * dy + dz * dz;
      float nd = fminf(dist[i], d);
      dist[i] = nd;
      if (nd > bv) { bv = nd; bi = i; }     // first occurrence: smallest i wins
    }
    rv[t] = bv; ri[t] = bi;
    __syncthreads();
    for (int off = 128; off > 0; off >>= 1) {
      if (t < off) {
        if (rv[t + off] > rv[t] || (rv[t + off] == rv[t] && ri[t + off] < ri[t])) {
          rv[t] = rv[t + off]; ri[t] = ri[t + off];
        }
      }
      __syncthreads();
    }
    if (t == 0) cur = ri[0];
    __syncthreads();
  }
}

// ------------------------------ gather centers -----------------------------
__global__ void gather_xyz(const float* __restrict__ xyz, const int* __restrict__ fidx,
                           float* __restrict__ nxyz, int Nin, int S) {
  int i = blockIdx.x * blockDim.x + threadIdx.x;     // over B*S
  if (i >= B_SZ * S) return;
  int b = i / S;
  int ci = fidx[i];
  const float* p = xyz + ((size_t)b * Nin + ci) * 3;
  nxyz[(size_t)i * 3 + 0] = p[0];
  nxyz[(size_t)i * 3 + 1] = p[1];
  nxyz[(size_t)i * 3 + 2] = p[2];
}

// -------------------------------- ball query -------------------------------
__global__ void ball_query(const float* __restrict__ xyz, const int* __restrict__ fidx,
                           int* __restrict__ gidx, int Nin, int S, float r2) {
  int i = blockIdx.x * blockDim.x + threadIdx.x;     // over B*S
  if (i >= B_SZ * S) return;
  int b = i / S;
  int ci = fidx[i];
  const float* base = xyz + (size_t)b * Nin * 3;
  float cx = base[ci * 3 + 0], cy = base[ci * 3 + 1], cz = base[ci * 3 + 2];
  int cnt = 0, first = -1;
  for (int n = 0; n < Nin && cnt < 32; ++n) {
    float dx = base[n * 3 + 0] - cx;
    float dy = base[n * 3 + 1] - cy;
    float dz = base[n * 3 + 2] - cz;
    float d = dx * dx + dy * dy + dz * dz;
    if (d <= r2) {                 // ascending in-radius indices, like reference
      if (first < 0) first = n;
      gidx[(size_t)i * 32 + cnt] = n;
      ++cnt;
    }
  }
  for (; cnt < 32; ++cnt) gidx[(size_t)i * 32 + cnt] = first;
}

// --------------- software-pipelined dual-chain WMMA K loop -----------------
// xr/wr point at this lane's A-row / B-col in float2 units; K2 % 4 == 0, >= 8.
__device__ inline v8f wmma_k_loop(const v2f* __restrict__ xr, const v2f* __restrict__ wr,
                                  int K2, int half) {
  v8f acc0 = {}, acc1 = {};
  v2f a0 = xr[half],     w0 = wr[half];
  v2f a1 = xr[2 + half], w1 = wr[2 + half];
  int k0 = 0;
  for (; k0 < K2 - 4; k0 += 4) {
    v2f na0 = xr[k0 + 4 + half], nw0 = wr[k0 + 4 + half];
    v2f na1 = xr[k0 + 6 + half], nw1 = wr[k0 + 6 + half];
    acc0 = __builtin_amdgcn_wmma_f32_16x16x4_f32(false, a0, false, w0, (short)0,
                                                 acc0, false, false);
    acc1 = __builtin_amdgcn_wmma_f32_16x16x4_f32(false, a1, false, w1, (short)0,
                                                 acc1, false, false);
    a0 = na0; w0 = nw0; a1 = na1; w1 = nw1;
  }
  acc0 = __builtin_amdgcn_wmma_f32_16x16x4_f32(false, a0, false, w0, (short)0,
                                               acc0, false, false);
  acc1 = __builtin_amdgcn_wmma_f32_16x16x4_f32(false, a1, false, w1, (short)0,
                                               acc1, false, false);
  return acc0 + acc1;
}

// --------------------- WMMA fp32 16x16x4 layer primitive -------------------
// One wave: 16 rows x Cout cols from LDS src, packed weights from global.
template <int CMAX>
__device__ inline void sa_layer(const float* __restrict__ src, float* __restrict__ dst,
                                const float* __restrict__ Wp, const float* __restrict__ bp,
                                int Cinp, int Cout, int lane, int wave) {
  int l16 = lane & 15;
  int half = lane >> 4;
  int m0 = wave * 16;
  const v2f* xr = (const v2f*)(src + (size_t)(m0 + l16) * CMAX);
  int K2 = Cinp >> 1;                       // float2 units; multiple of 4
  for (int n0 = 0; n0 < Cout; n0 += 16) {
    int n = n0 + l16;
    const v2f* wr = (const v2f*)(Wp + (size_t)n * Cinp);
    v8f acc = wmma_k_loop(xr, wr, K2, half);
    float bias = bp[n];
#pragma unroll
    for (int g = 0; g < 8; ++g) {
      float v = acc[g] + bias;
      dst[(size_t)(m0 + g + half * 8) * CMAX + n] = v > 0.0f ? v : 0.0f;
    }
  }
}

// ------------------ fused SA: group + 3-layer MLP + maxpool ----------------
template <int CIN, int C1, int C2, int C3, int CMAX>
__global__ __launch_bounds__(64) void sa_fused(const float* __restrict__ xyz,
                                               const float* __restrict__ pts, int Cpts,
                                               const int* __restrict__ fidx,
                                               const int* __restrict__ gidx,
                                               SAPp P, float* __restrict__ feat,
                                               int Nin, int S) {
  constexpr int CINP = (CIN + 7) & ~7;
  __shared__ __align__(16) float bufA[32 * CMAX];
  __shared__ __align__(16) float bufB[32 * CMAX];
  int gs = blockIdx.x;             // b*S + s
  int b = gs / S;
  int t = threadIdx.x;             // 64 threads = 2 waves
  int lane = t & 31, wave = t >> 5;

  for (int i = t; i < 32 * CMAX; i += 64) bufA[i] = 0.0f;
  __syncthreads();

  int ci = fidx[gs];
  const float* xb = xyz + (size_t)b * Nin * 3;
  float cx = xb[ci * 3 + 0], cy = xb[ci * 3 + 1], cz = xb[ci * 3 + 2];
  const int* gi = gidx + (size_t)gs * 32;
  for (int e = t; e < 32 * CIN; e += 64) {
    int k = e / CIN, c = e % CIN;
    int idx = gi[k];
    float v;
    if (c < 3) {
      float cc = (c == 0) ? cx : (c == 1) ? cy : cz;
      v = xb[idx * 3 + c] - cc;
    } else {
      v = pts[((size_t)b * Nin + idx) * Cpts + (c - 3)];
    }
    bufA[k * CMAX + c] = v;
  }
  __syncthreads();

  sa_layer<CMAX>(bufA, bufB, P.W0, P.b0, CINP, C1, lane, wave);
  __syncthreads();
  sa_layer<CMAX>(bufB, bufA, P.W1, P.b1, C1, C2, lane, wave);
  __syncthreads();
  sa_layer<CMAX>(bufA, bufB, P.W2, P.b2, C2, C3, lane, wave);
  __syncthreads();

  for (int c = t; c < C3; c += 64) {
    float m = bufB[c];
#pragma unroll 4
    for (int k = 1; k < 32; ++k) m = fmaxf(m, bufB[k * CMAX + c]);
    feat[(size_t)gs * C3 + c] = m;
  }
}

// ------------------------- generic row GEMM (WMMA) -------------------------
// Y[rows,Cout] = relu(X[rows,ldx] @ Wp^T + bp). rows % 32 == 0, Cinp % 8 == 0.
__global__ __launch_bounds__(64) void mlp_gemm(const float* __restrict__ X, int ldx,
                                               const float* __restrict__ Wp,
                                               const float* __restrict__ bp,
                                               float* __restrict__ Y,
                                               int Cinp, int Cout) {
  int lane = threadIdx.x & 31, wave = threadIdx.x >> 5;
  int l16 = lane & 15, half = lane >> 4;
  int mrow = blockIdx.x * 32 + wave * 16;
  int n = blockIdx.y * 16 + l16;
  const v2f* xr = (const v2f*)(X + (size_t)(mrow + l16) * ldx);
  const v2f* wr = (const v2f*)(Wp + (size_t)n * Cinp);
  __builtin_prefetch(X + (size_t)(mrow + l16) * ldx, 0, 1);
  int K2 = Cinp >> 1;                       // float2 units; multiple of 4
  v8f acc = wmma_k_loop(xr, wr, K2, half);
  float bias = bp[n];
#pragma unroll
  for (int g = 0; g < 8; ++g) {
    float v = acc[g] + bias;
    Y[(size_t)(mrow + g + half * 8) * Cout + n] = v > 0.0f ? v : 0.0f;
  }
}

// ----------------------- feature propagation helpers -----------------------
__global__ void copy_p1(const float* __restrict__ p1, float* __restrict__ cat,
                        int ld, int C1, int total) {
  int e = blockIdx.x * blockDim.x + threadIdx.x;
  if (e >= total) return;
  int r = e / C1, c = e % C1;
  cat[(size_t)r * ld + c] = p1[e];
}

__global__ void three_interp(const float* __restrict__ x1, const float* __restrict__ x2,
                             const float* __restrict__ p2, float* __restrict__ cat,
                             int ld, int coff, int N1, int N2, int C2) {
  int i = blockIdx.x * blockDim.x + threadIdx.x;     // over B*N1
  if (i >= B_SZ * N1) return;
  int b = i / N1;
  float px = x1[(size_t)i * 3 + 0], py = x1[(size_t)i * 3 + 1], pz = x1[(size_t)i * 3 + 2];
  const float* base = x2 + (size_t)b * N2 * 3;
  float d0 = 3.0e38f, d1 = 3.0e38f, d2 = 3.0e38f;
  int i0 = 0, i1 = 0, i2 = 0;
  for (int n = 0; n < N2; ++n) {
    float dx = base[n * 3 + 0] - px;
    float dy = base[n * 3 + 1] - py;
    float dz = base[n * 3 + 2] - pz;
    float d = dx * dx + dy * dy + dz * dz;
    if (d < d0)      { d2 = d1; i2 = i1; d1 = d0; i1 = i0; d0 = d; i0 = n; }
    else if (d < d1) { d2 = d1; i2 = i1; d1 = d;  i1 = n; }
    else if (d < d2) { d2 = d;  i2 = n; }
  }
  float w0 = 1.0f / (d0 + 1e-8f);
  float w1 = 1.0f / (d1 + 1e-8f);
  float w2 = 1.0f / (d2 + 1e-8f);
  float ws = w0 + w1 + w2;
  w0 /= ws; w1 /= ws; w2 /= ws;
  const float* pb = p2 + (size_t)b * N2 * C2;
  float* dst = cat + (size_t)i * ld + coff;
  for (int c = 0; c < C2; ++c)
    dst[c] = w0 * pb[(size_t)i0 * C2 + c] + w1 * pb[(size_t)i1 * C2 + c] +
             w2 * pb[(size_t)i2 * C2 + c];
}

// --------------------------------- head2 -----------------------------------
__global__ void head2_kernel(const float* __restrict__ X, const float* __restrict__ W,
                             const float* __restrict__ bias, float* __restrict__ out,
                             int rows) {
  int r = blockIdx.x * blockDim.x + threadIdx.x;
  if (r >= rows) return;
  float a0 = bias[0], a1 = bias[1];
  const float* x = X + (size_t)r * 128;
  for (int c = 0; c < 128; ++c) {
    float v = x[c];
    a0 += v * W[c];
    a1 += v * W[128 + c];
  }
  out[(size_t)r * 2 + 0] = a0;
  out[(size_t)r * 2 + 1] = a1;
}

// ============================== host driver ================================
extern "C" void kernel_launch(void* const* d_in, const int* in_sizes, int n_in,
                              void* d_out, int out_size, void* d_ws, size_t ws_size,
                              hipStream_t stream) {
  const float inv = 1.0f / sqrtf(1.0f + 1e-5f);
  const float* xin = (const float*)d_in[0];

  int p = 1;
  auto nextL = [&](LayerP& L) {
    L.W  = (const float*)d_in[p + 0];
    L.b  = (const float*)d_in[p + 1];
    L.g  = (const float*)d_in[p + 2];
    L.be = (const float*)d_in[p + 3];
    p += 4;
  };
  LayerP saL[4][3];
  for (int i = 0; i < 4; ++i)
    for (int l = 0; l < 3; ++l) nextL(saL[i][l]);
  LayerP fp4L[2]; nextL(fp4L[0]); nextL(fp4L[1]);
  LayerP fp3L[2]; nextL(fp3L[0]); nextL(fp3L[1]);
  LayerP fp2L[2]; nextL(fp2L[0]); nextL(fp2L[1]);
  LayerP fp1L[3]; nextL(fp1L[0]); nextL(fp1L[1]); nextL(fp1L[2]);
  LayerP h1; nextL(h1);
  const float* W2 = (const float*)d_in[p++];
  const float* b2 = (const float*)d_in[p++];

  // workspace carve (fp32 elements, 256B aligned)
  char* base = (char*)d_ws;
  size_t off = 0;
  auto alloc = [&](size_t elems) -> float* {
    float* r = (float*)(base + off);
    off += ((elems * 4 + 255) / 256) * 256;
    return r;
  };
  float* xyz0 = alloc((size_t)B_SZ * N0 * 3);
  float* pts0 = alloc((size_t)B_SZ * N0 * 6);
  float* xyz1 = alloc((size_t)B_SZ * 1024 * 3);
  float* xyz2 = alloc((size_t)B_SZ * 256 * 3);
  float* xyz3 = alloc((size_t)B_SZ * 64 * 3);
  float* xyz4 = alloc((size_t)B_SZ * 16 * 3);
  int* fidx1 = (int*)alloc((size_t)B_SZ * 1024);
  int* fidx2 = (int*)alloc((size_t)B_SZ * 256);
  int* fidx3 = (int*)alloc((size_t)B_SZ * 64);
  int* fidx4 = (int*)alloc((size_t)B_SZ * 16);
  int* gidx1 = (int*)alloc((size_t)B_SZ * 1024 * 32);
  int* gidx2 = (int*)alloc((size_t)B_SZ * 256 * 32);
  int* gidx3 = (int*)alloc((size_t)B_SZ * 64 * 32);
  int* gidx4 = (int*)alloc((size_t)B_SZ * 16 * 32);
  float* feat1 = alloc((size_t)B_SZ * 1024 * 64);
  float* feat2 = alloc((size_t)B_SZ * 256 * 128);
  float* feat3 = alloc((size_t)B_SZ * 64 * 256);
  float* feat4 = alloc((size_t)B_SZ * 16 * 512);
  float* cat4 = alloc((size_t)B_SZ * 64 * 768);
  float* cat3 = alloc((size_t)B_SZ * 256 * 384);
  float* cat2 = alloc((size_t)B_SZ * 1024 * 320);
  float* cat0 = alloc((size_t)B_SZ * 4096 * 136);     // ld=136 (134 padded, %8==0)
  float* fpo3 = alloc((size_t)B_SZ * 64 * 256);
  float* fpo2 = alloc((size_t)B_SZ * 256 * 256);
  float* fpo1 = alloc((size_t)B_SZ * 1024 * 128);
  float* fpo0 = alloc((size_t)B_SZ * 4096 * 128);
  float* actA = alloc((size_t)65536 * 128);
  float* actB = alloc((size_t)65536 * 128);

  // ---- pack all MLP weights (BN folded, K zero-padded to mult of 8) ----
  struct PW { const float *W, *b; int Cinp; };
  auto packL = [&](const LayerP& L, int Cout, int Cin) -> PW {
    int Cinp = (Cin + 7) & ~7;
    float* Wp = alloc((size_t)Cout * Cinp);
    float* bp = alloc(Cout);
    int total = Cout * Cinp;
    pack_weight<<<(total + 255) / 256, 256, 0, stream>>>(L.W, L.g, Wp, Cin, Cinp, total, inv);
    pack_bias<<<(Cout + 63) / 64, 64, 0, stream>>>(L.b, L.g, L.be, bp, Cout, inv);
    return PW{Wp, bp, Cinp};
  };
  const int saCin[4][3]  = {{9, 32, 32}, {67, 64, 64}, {131, 128, 128}, {259, 256, 256}};
  const int saCout[4][3] = {{32, 32, 64}, {64, 64, 128}, {128, 128, 256}, {256, 256, 512}};
  PW saP[4][3];
  for (int i = 0; i < 4; ++i)
    for (int l = 0; l < 3; ++l) saP[i][l] = packL(saL[i][l], saCout[i][l], saCin[i][l]);
  PW fp4P[2] = {packL(fp4L[0], 256, 768), packL(fp4L[1], 256, 256)};
  PW fp3P[2] = {packL(fp3L[0], 256, 384), packL(fp3L[1], 256, 256)};
  PW fp2P[2] = {packL(fp2L[0], 256, 320), packL(fp2L[1], 128, 256)};
  PW fp1P[3] = {packL(fp1L[0], 128, 134), packL(fp1L[1], 128, 128), packL(fp1L[2], 128, 128)};
  PW h1P = packL(h1, 128, 128);

  auto gemm = [&](const float* X, int ldx, const PW& L, float* Y, int rows, int Cout) {
    dim3 g(rows / 32, Cout / 16);
    mlp_gemm<<<g, 64, 0, stream>>>(X, ldx, L.W, L.b, Y, L.Cinp, Cout);
  };
  auto mkSAP = [&](PW* a) {
    return SAPp{a[0].W, a[0].b, a[1].W, a[1].b, a[2].W, a[2].b};
  };

  // ------------- input split -------------
  split_input<<<(B_SZ * N0 + 255) / 256, 256, 0, stream>>>(xin, xyz0, pts0);

  // ------------- SA level 1: 4096 -> 1024, r=0.1, Cin=9, mlp 32/32/64 -------------
  fps_kernel<<<B_SZ, 256, 0, stream>>>(xyz0, fidx1, 4096, 1024);
  ball_query<<<(B_SZ * 1024 + 255) / 256, 256, 0, stream>>>(xyz0, fidx1, gidx1, 4096, 1024, 0.01f);
  gather_xyz<<<(B_SZ * 1024 + 255) / 256, 256, 0, stream>>>(xyz0, fidx1, xyz1, 4096, 1024);
  sa_fused<9, 32, 32, 64, 64><<<B_SZ * 1024, 64, 0, stream>>>(
      xyz0, pts0, 6, fidx1, gidx1, mkSAP(saP[0]), feat1, 4096, 1024);

  // ------------- SA level 2: 1024 -> 256, r=0.2, Cin=67, mlp 64/64/128 -------------
  fps_kernel<<<B_SZ, 256, 0, stream>>>(xyz1, fidx2, 1024, 256);
  ball_query<<<(B_SZ * 256 + 255) / 256, 256, 0, stream>>>(xyz1, fidx2, gidx2, 1024, 256, 0.04f);
  gather_xyz<<<(B_SZ * 256 + 255) / 256, 256, 0, stream>>>(xyz1, fidx2, xyz2, 1024, 256);
  sa_fused<67, 64, 64, 128, 128><<<B_SZ * 256, 64, 0, stream>>>(
      xyz1, feat1, 64, fidx2, gidx2, mkSAP(saP[1]), feat2, 1024, 256);

  // ------------- SA level 3: 256 -> 64, r=0.4, Cin=131, mlp 128/128/256 -------------
  fps_kernel<<<B_SZ, 256, 0, stream>>>(xyz2, fidx3, 256, 64);
  ball_query<<<(B_SZ * 64 + 255) / 256, 256, 0, stream>>>(xyz2, fidx3, gidx3, 256, 64, 0.16f);
  gather_xyz<<<(B_SZ * 64 + 255) / 256, 256, 0, stream>>>(xyz2, fidx3, xyz3, 256, 64);
  sa_fused<131, 128, 128, 256, 256><<<B_SZ * 64, 64, 0, stream>>>(
      xyz2, feat2, 128, fidx3, gidx3, mkSAP(saP[2]), feat3, 256, 64);

  // ------------- SA level 4: 64 -> 16, r=0.8, Cin=259, mlp 256/256/512 -------------
  fps_kernel<<<B_SZ, 256, 0, stream>>>(xyz3, fidx4, 64, 16);
  ball_query<<<(B_SZ * 16 + 255) / 256, 256, 0, stream>>>(xyz3, fidx4, gidx4, 64, 16, 0.64f);
  gather_xyz<<<(B_SZ * 16 + 255) / 256, 256, 0, stream>>>(xyz3, fidx4, xyz4, 64, 16);
  sa_fused<259, 256, 256, 512, 512><<<B_SZ * 16, 64, 0, stream>>>(
      xyz3, feat3, 256, fidx4, gidx4, mkSAP(saP[3]), feat4, 64, 16);

  // ------------- FP4: l4 -> l3, cat 256+512=768, mlp 256/256 -------------
  copy_p1<<<(B_SZ * 64 * 256 + 255) / 256, 256, 0, stream>>>(feat3, cat4, 768, 256, B_SZ * 64 * 256);
  three_interp<<<(B_SZ * 64 + 255) / 256, 256, 0, stream>>>(xyz3, xyz4, feat4, cat4, 768, 256, 64, 16, 512);
  gemm(cat4, 768, fp4P[0], actA, B_SZ * 64, 256);
  gemm(actA, 256, fp4P[1], fpo3, B_SZ * 64, 256);

  // ------------- FP3: l3 -> l2, cat 128+256=384, mlp 256/256 -------------
  copy_p1<<<(B_SZ * 256 * 128 + 255) / 256, 256, 0, stream>>>(feat2, cat3, 384, 128, B_SZ * 256 * 128);
  three_interp<<<(B_SZ * 256 + 255) / 256, 256, 0, stream>>>(xyz2, xyz3, fpo3, cat3, 384, 128, 256, 64, 256);
  gemm(cat3, 384, fp3P[0], actA, B_SZ * 256, 256);
  gemm(actA, 256, fp3P[1], fpo2, B_SZ * 256, 256);

  // ------------- FP2: l2 -> l1, cat 64+256=320, mlp 256/128 -------------
  copy_p1<<<(B_SZ * 1024 * 64 + 255) / 256, 256, 0, stream>>>(feat1, cat2, 320, 64, B_SZ * 1024 * 64);
  three_interp<<<(B_SZ * 1024 + 255) / 256, 256, 0, stream>>>(xyz1, xyz2, fpo2, cat2, 320, 64, 1024, 256, 256);
  gemm(cat2, 320, fp2P[0], actA, B_SZ * 1024, 256);
  gemm(actA, 256, fp2P[1], fpo1, B_SZ * 1024, 128);

  // ------------- FP1: l1 -> l0, cat 6+128=134 (ld 136), mlp 128/128/128 -------------
  copy_p1<<<(B_SZ * 4096 * 6 + 255) / 256, 256, 0, stream>>>(pts0, cat0, 136, 6, B_SZ * 4096 * 6);
  three_interp<<<(B_SZ * 4096 + 255) / 256, 256, 0, stream>>>(xyz0, xyz1, fpo1, cat0, 136, 6, 4096, 1024, 128);
  zero_pad_cols<<<(B_SZ * 4096 * 2 + 255) / 256, 256, 0, stream>>>(cat0, 136, 134, B_SZ * 4096);
  gemm(cat0, 136, fp1P[0], actA, B_SZ * 4096, 128);
  gemm(actA, 128, fp1P[1], actB, B_SZ * 4096, 128);
  gemm(actB, 128, fp1P[2], fpo0, B_SZ * 4096, 128);

  // ------------- head -------------
  gemm(fpo0, 128, h1P, actA, B_SZ * 4096, 128);
  head2_kernel<<<(B_SZ * 4096 + 255) / 256, 256, 0, stream>>>(actA, W2, b2, (float*)d_out, B_SZ * 4096);

  (void)in_sizes; (void)n_in; (void)out_size; (void)ws_size;
}